// MResConvPoint_58969900974235
// MI455X (gfx1250) — compile-verified
//
#include <hip/hip_runtime.h>
#include <hip/hip_bf16.h>

typedef float v2f __attribute__((ext_vector_type(2)));
typedef float v4f __attribute__((ext_vector_type(4)));
typedef float v8f __attribute__((ext_vector_type(8)));

#define NB   4
#define NCH  64
#define NV   100000
#define NK   6
#define VT   10          // vertex tiles (of 16) per block
#define GRIDX 625        // 6250 tiles / VT
#define GS   132         // G LDS row stride (floats): conflict-free b64 reads
#define HS   68          // H staging row stride (16B-aligned rows)
#define RS   68          // residual staging row stride

// ---------------- kernel 0: transpose x [B,C,N] -> xT [B,N,C] ----------------
__global__ void __launch_bounds__(256) k_transpose(const float* __restrict__ x,
                                                   float* __restrict__ xT) {
  __shared__ float t[64 * 33];
  const int b = blockIdx.y;
  const int n0 = blockIdx.x * 32;
  const int tx = threadIdx.x & 31;   // n within tile
  const int ty = threadIdx.x >> 5;   // 0..7
  const float* xb = x + (size_t)b * NCH * NV;
#pragma unroll
  for (int i = 0; i < 8; ++i) {
    int c = i * 8 + ty;
    t[c * 33 + tx] = xb[(size_t)c * NV + n0 + tx];   // coalesced along n
  }
  __syncthreads();
  float* xTb = xT + ((size_t)b * NV + n0) * NCH;
  const int c = threadIdx.x & 63;
  const int v0 = threadIdx.x >> 6;   // 0..3
#pragma unroll
  for (int i = 0; i < 8; ++i) {
    int v = i * 4 + v0;
    xTb[(size_t)v * NCH + c] = t[c * 33 + v];        // coalesced along c
  }
}

// -------- kernel 0b: pack W[o][c][s] -> Wcat[o][ccat] (ccat = s*64 + c) ------
__global__ void k_wcat(const float* __restrict__ W0, const float* __restrict__ W1,
                       float* __restrict__ Wc0, float* __restrict__ Wc1) {
  int t = blockIdx.x * 256 + threadIdx.x;
  if (t < 64 * 128) {
    int o = t >> 7, cc = t & 127;
    int s = cc >> 6, cin = cc & 63;
    Wc0[t] = W0[o * 128 + cin * 2 + s];
    Wc1[t] = W1[o * 128 + cin * 2 + s];
  }
}

// --------------- kernel 1: conv1 (WMMA) + BN stat partials -------------------
__global__ void __launch_bounds__(128) k_conv1(
    const float* __restrict__ xT, const int* __restrict__ neigh,
    const float* __restrict__ Wc, float* __restrict__ hT,
    float* __restrict__ stats) {
  __shared__ float G[16 * GS];   // [vertex][128 cat-channels], padded
  __shared__ float H[16 * HS];   // output staging for coalesced hT writes
  const int b = blockIdx.y;
  const int lane = threadIdx.x & 31;
  const int wv = threadIdx.x >> 5;      // wave 0..3 -> output-channel stripe
  const int o0 = wv * 16;
  const int vl = lane & 15;
  const int hf = lane >> 4;

  // A-fragments: W slice 16x128 resident in 64 VGPRs (loaded once per block)
  v2f w[32];
  {
    const float* wp = Wc + (o0 + vl) * 128 + 2 * hf;
#pragma unroll
    for (int k = 0; k < 32; ++k) w[k] = *(const v2f*)(wp + 4 * k);
  }
  const float* xTb = xT + (size_t)b * NV * NCH;
  const int* nbp = neigh + (size_t)b * NV * NK;

  float s0[8], s1[8];
#pragma unroll
  for (int r = 0; r < 8; ++r) { s0[r] = 0.f; s1[r] = 0.f; }

  const int gv = threadIdx.x >> 3;       // gather: vertex 0..15
  const int cb = (threadIdx.x & 7) * 8;  // gather: 8-channel slice

  for (int ti = 0; ti < VT; ++ti) {
    const int n0 = (blockIdx.x * VT + ti) * 16;
    // ---- build G tile: center row + sum of 6 neighbor rows (coalesced 256B rows)
    {
      const float* cr = xTb + (size_t)(n0 + gv) * NCH + cb;
      v4f c0 = *(const v4f*)cr;
      v4f c1 = *(const v4f*)(cr + 4);
      v4f a0 = {}, a1 = {};
      const int* ip = nbp + (size_t)(n0 + gv) * NK;
#pragma unroll
      for (int j = 0; j < NK; ++j) {
        const float* nr = xTb + (size_t)ip[j] * NCH + cb;
        a0 += *(const v4f*)nr;
        a1 += *(const v4f*)(nr + 4);
      }
      float* g = G + gv * GS;
      *(v4f*)(g + cb) = c0;
      *(v4f*)(g + cb + 4) = c1;
      *(v4f*)(g + 64 + cb) = a0;
      *(v4f*)(g + 64 + cb + 4) = a1;
      if (ti + 1 < VT)
        __builtin_prefetch(xTb + (size_t)(n0 + 16 + gv) * NCH + cb, 0, 3);
    }
    __syncthreads();
    // ---- 32 chained f32 WMMAs: [16x4]x[4x16] over K=128
    v8f acc = {};
    {
      const float* gB = G + vl * GS + 2 * hf;   // conflict-free ds_load_b64
#pragma unroll
      for (int k = 0; k < 32; ++k) {
        v2f bf = *(const v2f*)(gB + 4 * k);
        acc = __builtin_amdgcn_wmma_f32_16x16x4_f32(
            false, w[k], false, bf, (short)0, acc, false, false);
      }
    }
    // ---- BN stat partials on relu(h); stage raw h for coalesced store
#pragma unroll
    for (int r = 0; r < 8; ++r) {
      float v = acc[r];
      float rv = fmaxf(v, 0.f);
      s0[r] += rv;
      s1[r] += rv * rv;
      H[vl * HS + o0 + r + 8 * hf] = v;
    }
    __syncthreads();
    {
      float* hr = hT + ((size_t)b * NV + n0 + gv) * NCH + cb;
      const float* hs = H + gv * HS + cb;
      *(v4f*)hr = *(const v4f*)hs;
      *(v4f*)(hr + 4) = *(const v4f*)(hs + 4);
    }
  }
  // ---- reduce stats within each 16-lane group, atomically add to 32 slots
#pragma unroll
  for (int r = 0; r < 8; ++r) {
    float a = s0[r], q = s1[r];
#pragma unroll
    for (int m = 8; m >= 1; m >>= 1) {
      a += __shfl_xor(a, m);
      q += __shfl_xor(q, m);
    }
    if (vl == 0) {
      int ch = o0 + r + 8 * hf;
      int slot = (blockIdx.x + blockIdx.y * 13) & 31;
      atomicAdd(stats + slot * 128 + ch, a);
      atomicAdd(stats + slot * 128 + 64 + ch, q);
    }
  }
}

// ------------- kernel 2: finalize BN -> per-channel scale/shift --------------
__global__ void k_bn(const float* __restrict__ stats, const float* __restrict__ gamma,
                     const float* __restrict__ beta, float* __restrict__ bn) {
  int c = threadIdx.x;  // 64 threads
  float s = 0.f, q = 0.f;
  for (int i = 0; i < 32; ++i) {
    s += stats[i * 128 + c];
    q += stats[i * 128 + 64 + c];
  }
  const float inv = 1.0f / (float)(NB * NV);
  float mean = s * inv;
  float var = q * inv - mean * mean;
  float sc = gamma[c] * rsqrtf(var + 1e-5f);
  bn[c] = sc;
  bn[64 + c] = beta[c] - mean * sc;
}

// ---- kernel 3: conv2 (WMMA) with fused BN(relu) on inputs + residual + relu --
__global__ void __launch_bounds__(128) k_conv2(
    const float* __restrict__ hT, const int* __restrict__ neigh,
    const float* __restrict__ Wc, const float* __restrict__ bn,
    float* __restrict__ out) {
  __shared__ float G[16 * GS];
  __shared__ float R[16 * RS];   // raw center h for residual
  const int b = blockIdx.y;
  const int lane = threadIdx.x & 31;
  const int wv = threadIdx.x >> 5;
  const int o0 = wv * 16;
  const int vl = lane & 15;
  const int hf = lane >> 4;

  v2f w[32];
  {
    const float* wp = Wc + (o0 + vl) * 128 + 2 * hf;
#pragma unroll
    for (int k = 0; k < 32; ++k) w[k] = *(const v2f*)(wp + 4 * k);
  }
  const float* hTb = hT + (size_t)b * NV * NCH;
  const int* nbp = neigh + (size_t)b * NV * NK;

  const int gv = threadIdx.x >> 3;
  const int cb = (threadIdx.x & 7) * 8;
  const v4f scA = *(const v4f*)(bn + cb);
  const v4f scB = *(const v4f*)(bn + cb + 4);
  const v4f shA = *(const v4f*)(bn + 64 + cb);
  const v4f shB = *(const v4f*)(bn + 64 + cb + 4);

  for (int ti = 0; ti < VT; ++ti) {
    const int n0 = (blockIdx.x * VT + ti) * 16;
    {
      const float* cr = hTb + (size_t)(n0 + gv) * NCH + cb;
      v4f c0 = *(const v4f*)cr;
      v4f c1 = *(const v4f*)(cr + 4);
      const v4f z = {};
      v4f f0 = __builtin_elementwise_max(c0, z) * scA + shA;  // BN(relu(center))
      v4f f1 = __builtin_elementwise_max(c1, z) * scB + shB;
      v4f a0 = {}, a1 = {};
      const int* ip = nbp + (size_t)(n0 + gv) * NK;
#pragma unroll
      for (int j = 0; j < NK; ++j) {
        const float* nr = hTb + (size_t)ip[j] * NCH + cb;
        a0 += __builtin_elementwise_max(*(const v4f*)nr, z);
        a1 += __builtin_elementwise_max(*(const v4f*)(nr + 4), z);
      }
      a0 = a0 * scA + 6.f * shA;  // sum_j BN(relu(h_j))
      a1 = a1 * scB + 6.f * shB;
      float* g = G + gv * GS;
      *(v4f*)(g + cb) = f0;
      *(v4f*)(g + cb + 4) = f1;
      *(v4f*)(g + 64 + cb) = a0;
      *(v4f*)(g + 64 + cb + 4) = a1;
      float* rr = R + gv * RS;
      *(v4f*)(rr + cb) = c0;          // raw residual
      *(v4f*)(rr + cb + 4) = c1;
      if (ti + 1 < VT)
        __builtin_prefetch(hTb + (size_t)(n0 + 16 + gv) * NCH + cb, 0, 3);
    }
    __syncthreads();
    v8f acc = {};
    {
      const float* gB = G + vl * GS + 2 * hf;
#pragma unroll
      for (int k = 0; k < 32; ++k) {
        v2f bf = *(const v2f*)(gB + 4 * k);
        acc = __builtin_amdgcn_wmma_f32_16x16x4_f32(
            false, w[k], false, bf, (short)0, acc, false, false);
      }
    }
#pragma unroll
    for (int r = 0; r < 8; ++r) {
      int o = o0 + r + 8 * hf;
      float v = acc[r] + R[vl * RS + o];        // + residual
      out[((size_t)b * NCH + o) * NV + n0 + vl] = fmaxf(v, 0.f);
    }
    __syncthreads();
  }
}

// ---------------------------------------------------------------------------
extern "C" void kernel_launch(void* const* d_in, const int* in_sizes, int n_in,
                              void* d_out, int out_size, void* d_ws, size_t ws_size,
                              hipStream_t stream) {
  const float* x     = (const float*)d_in[0];
  const int*   neigh = (const int*)d_in[1];
  const float* W0    = (const float*)d_in[2];
  const float* W1    = (const float*)d_in[3];
  const float* gamma = (const float*)d_in[4];
  const float* beta  = (const float*)d_in[5];
  float* out = (float*)d_out;

  const size_t XE = (size_t)NB * NV * NCH;     // 25,600,000 floats
  float* fws   = (float*)d_ws;
  float* xT    = fws;
  float* hT    = fws + XE;
  float* Wc0   = fws + 2 * XE;
  float* Wc1   = Wc0 + 64 * 128;
  float* stats = Wc1 + 64 * 128;
  float* bn    = stats + 32 * 128;
  const size_t need = (2 * XE + 2 * 64 * 128 + 32 * 128 + 128) * sizeof(float);
  if (ws_size < need) return;  // ~205 MB scratch required

  hipMemsetAsync(stats, 0, 32 * 128 * sizeof(float), stream);
  k_transpose<<<dim3(NV / 32, NB), 256, 0, stream>>>(x, xT);
  k_wcat<<<32, 256, 0, stream>>>(W0, W1, Wc0, Wc1);
  k_conv1<<<dim3(GRIDX, NB), 128, 0, stream>>>(xT, neigh, Wc0, hT, stats);
  k_bn<<<1, 64, 0, stream>>>(stats, gamma, beta, bn);
  k_conv2<<<dim3(GRIDX, NB), 128, 0, stream>>>(hT, neigh, Wc1, bn, out);
}